// general_EPIs_dense_34248069218806
// MI455X (gfx1250) — compile-verified
//
#include <hip/hip_runtime.h>

#define BB 2
#define KK 3
#define NN 4096
#define TSZ 4
#define BKN (BB*KK*NN)
#define PSPLIT 8                 // p-dimension split across blocks
#define NCH (NN/64/PSPLIT)       // 64-wide p-chunks per block = 8

typedef __attribute__((ext_vector_type(2))) float v2f;
typedef __attribute__((ext_vector_type(8))) float v8f;

// workspace layout in floats
#define WS_LINC   0          // 576 floats: linC[j][i][c][a][m]
#define WS_PSTS   1024       // B*K*N*16 = 393216 floats (normalized Psts)
#define WS_W      394240     // B*K*N = 24576 floats (i_ * p10)
#define WS_PART   443392     // PSPLIT * B*K*N = 196608 floats (partial products)

// --- CDNA5 async global->LDS path (guarded; falls back to sync loads) ------
#if __has_builtin(__builtin_amdgcn_global_load_async_to_lds_b32) && \
    __has_builtin(__builtin_amdgcn_s_wait_asynccnt)
#define USE_ASYNC_LDS 1
typedef __attribute__((address_space(1))) int GInt;   // global int
typedef __attribute__((address_space(3))) int LInt;   // LDS int
#else
#define USE_ASYNC_LDS 0
#endif

// ---------------------------------------------------------------------------
// Phase 0: build linC[j,i,c,a,m] (tiny, single thread)
// ---------------------------------------------------------------------------
__global__ void prep_kernel(const float* __restrict__ ps,    // [1,K,4,4]
                            const float* __restrict__ cim,   // [K,K]
                            const float* __restrict__ Ts,    // [1,K]
                            const float* __restrict__ trust, // [1,K]
                            const float* __restrict__ tt,    // [1]
                            const int*   __restrict__ cat,   // [1]
                            float* __restrict__ linc) {
  if (threadIdx.x != 0 || blockIdx.x != 0) return;
  const int category = cat[0];
  const float mask1[4][4] = {{0.f,0.f,-100.f,-100.f},
                             {-100.f,0.f,0.f,-100.f},
                             {-100.f,-100.f,0.f,0.f},
                             {-100.f,-100.f,-100.f,0.f}};
  float psm[KK][4][4];
  for (int i = 0; i < KK; ++i)
    for (int c = 0; c < 4; ++c) {
      float v[4], m = -1e30f, s = 0.f;
      for (int a = 0; a < 4; ++a) {
        v[a] = ps[((category*KK + i)*4 + c)*4 + a] + mask1[c][a];
        m = fmaxf(m, v[a]);
      }
      for (int a = 0; a < 4; ++a) { v[a] = expf(v[a] - m); s += v[a]; }
      for (int a = 0; a < 4; ++a) psm[i][c][a] = v[a] / s;
    }
  const float tv = tt[0];
  float sig[KK];
  for (int i = 0; i < KK; ++i)
    sig[i] = 1.0f / (1.0f + expf(-trust[category*KK + i] * (tv - Ts[category*KK + i])));
  for (int j = 0; j < KK; ++j)
    for (int i = 0; i < KK; ++i)
      for (int c = 0; c < 4; ++c)
        for (int a = 0; a < 4; ++a) {
          const float cl = (c == a) ? 1.0f : sig[i];
          for (int m = 0; m < 4; ++m) {
            float lv;
            if (j == i)                          lv = psm[i][a][c];          // diag: psMatrix[i].T
            else if (c == 1 && a == 0 && m >= 2) lv = 0.5f*(cim[j*KK+i] + cim[i*KK+j]);
            else                                 lv = 1.0f;
            linc[((j*KK + i)*16 + c*4 + a)*4 + m] = lv * cl;
          }
        }
}

// ---------------------------------------------------------------------------
// Phase 1: Psts via WMMA f32 16x16x4 (one wave per (b, 16-wide p-tile))
// ---------------------------------------------------------------------------
__global__ void psts_kernel(const float* __restrict__ state,  // [B,K,N,4]
                            const float* __restrict__ linc,   // [j][i][16][4]
                            float* __restrict__ psts,         // [bk][p][16]
                            float* __restrict__ wout) {       // [bk][p]
  const int lane = threadIdx.x;                 // 0..31 (wave32)
  const int blk  = blockIdx.x;
  const int b    = blk >> 8;
  const int p0   = (blk & 255) << 4;
  const int col  = lane & 15;
  const int hi   = lane >> 4;
  const int kb   = hi * 2;

  v2f Bm[KK];
  for (int j = 0; j < KK; ++j) {
    const float* sp = state + (((size_t)(b*KK + j)*NN + (p0 + col))*TSZ);
    Bm[j].x = sp[kb];
    Bm[j].y = sp[kb + 1];
  }
  const int row = lane & 15;
  const int c_  = row >> 2, a_ = row & 3;

  float p10v[KK];
  v8f   accs[KK];

  for (int i = 0; i < KK; ++i) {
    v8f acc;
    for (int j = 0; j < KK; ++j) {
      const float* lp = linc + ((j*KK + i)*16 + c_*4 + a_)*4 + kb;
      v2f Am; Am.x = lp[0]; Am.y = lp[1];
      v8f zero = {0.f,0.f,0.f,0.f,0.f,0.f,0.f,0.f};
      v8f d = __builtin_amdgcn_wmma_f32_16x16x4_f32(
                  false, Am, false, Bm[j], (short)0, zero, false, false);
      acc = (j == 0) ? d : (acc * d);
    }
    for (int a = 0; a < 4; ++a) {
      float s = acc[a] + acc[a + 4];
      s += __shfl_xor(s, 16, 32);
      s += 1e-8f;
      acc[a]     = acc[a]     / s;
      acc[a + 4] = acc[a + 4] / s;
    }
    accs[i] = acc;
    p10v[i] = acc[4];
  }

  for (int i = 0; i < KK; ++i) {
    float* pp = psts + ((size_t)(b*KK + i)*NN + (p0 + col))*16;
    v8f acc = accs[i];
    for (int e = 0; e < 8; ++e) pp[e + hi*8] = acc[e];
  }
  if (hi == 0) {
    for (int i = 0; i < KK; ++i) {
      const float iv = state[(((size_t)(b*KK + i)*NN + (p0 + col))*TSZ) + 2];
      wout[(b*KK + i)*NN + p0 + col] = p10v[i] * iv;
    }
  }
}

// ---------------------------------------------------------------------------
// Phase 2: partial prodP over a 512-wide p segment for a 64-wide q tile.
// Double-buffered tiles via async global->LDS B32 DMA (per-lane LDS scatter
// lets us transpose contact[q][p] into BtT[pl][ql] during the copy).
// 34 uniform async ops per thread per chunk -> s_wait_asynccnt 34 pipelining.
// Layout [64][65]: all LDS reads/writes are conflict-free, 4B-aligned.
// ---------------------------------------------------------------------------
__device__ __forceinline__ void issue_chunk(float (* __restrict__ At)[65],   // At[pl][ql]
                                            float (* __restrict__ BtT)[65],  // BtT[pl][ql]
                                            float* __restrict__ wflat,       // [384]
                                            const float* __restrict__ contact,
                                            const float* __restrict__ wv,
                                            int p0, int q0, int t) {
#if USE_ASYNC_LDS
  for (int k = 0; k < 16; ++k) {
    const int idx = t + k*256;               // 0..4095
    const int r = idx >> 6, c = idx & 63;
    // At[r][c] = contact[p0+r][q0+c]   (contiguous in c -> coalesced)
    __builtin_amdgcn_global_load_async_to_lds_b32(
        (GInt*)(contact + (size_t)(p0 + r)*NN + (q0 + c)),
        (LInt*)&At[r][c], 0, 0);
  }
  for (int k = 0; k < 16; ++k) {
    const int idx = t + k*256;
    const int r = idx >> 6, c = idx & 63;
    // BtT[c][r] = contact[q0+r][p0+c]  (coalesced read, LDS scatter-transpose)
    __builtin_amdgcn_global_load_async_to_lds_b32(
        (GInt*)(contact + (size_t)(q0 + r)*NN + (p0 + c)),
        (LInt*)&BtT[c][r], 0, 0);
  }
  for (int k = 0; k < 2; ++k) {
    const int f  = (t + k*256) % 384;        // wrap -> uniform count per wave
    const int bk = f >> 6, pl = f & 63;
    __builtin_amdgcn_global_load_async_to_lds_b32(
        (GInt*)(wv + bk*NN + p0 + pl),
        (LInt*)&wflat[f], 0, 0);
  }
#else
  for (int idx = t; idx < 4096; idx += 256) {
    const int r = idx >> 6, c = idx & 63;
    At[r][c]  = contact[(size_t)(p0 + r)*NN + (q0 + c)];
    BtT[c][r] = contact[(size_t)(q0 + r)*NN + (p0 + c)];
  }
  for (int idx = t; idx < 384; idx += 256) {
    const int bk = idx >> 6, pl = idx & 63;
    wflat[idx] = wv[bk*NN + p0 + pl];
  }
#endif
}

__device__ __forceinline__ void wait_chunk(bool more_inflight) {
#if USE_ASYNC_LDS
  if (more_inflight) __builtin_amdgcn_s_wait_asynccnt(34);
  else               __builtin_amdgcn_s_wait_asynccnt(0);
#endif
  __syncthreads();
}

__global__ void prod_kernel(const float* __restrict__ contact, // [N,N]
                            const float* __restrict__ state,   // [B,K,N,4]
                            const float* __restrict__ wv,      // [bk][p]
                            float* __restrict__ part) {        // [pseg][bk][q]
  __shared__ float At[2][64][65];
  __shared__ float BtT[2][64][65];
  __shared__ float wch[2][384];
  __shared__ float red[6][4][64];
  const int t    = threadIdx.x;               // 0..255
  const int qt   = blockIdx.x & 63;
  const int pseg = blockIdx.x >> 6;           // 0..PSPLIT-1
  const int q0   = qt * 64;
  const int ql   = t & 63;
  const int psub = t >> 6;                    // 0..3

  float sreg[6], prod[6];
  for (int bk = 0; bk < 6; ++bk) {
    sreg[bk] = state[((size_t)bk*NN + (q0 + ql))*TSZ + 0];
    prod[bk] = 1.0f;
  }

  issue_chunk(At[0], BtT[0], wch[0], contact, wv, (pseg*NCH + 0)*64, q0, t);

  for (int pc = 0; pc < NCH; ++pc) {
    const int cur = pc & 1;
    if (pc + 1 < NCH)
      issue_chunk(At[cur^1], BtT[cur^1], wch[cur^1], contact, wv,
                  (pseg*NCH + pc + 1)*64, q0, t);
    wait_chunk(pc + 1 < NCH);

    const int pb = psub * 16;
    for (int pl = pb; pl < pb + 16; ++pl) {
      const float cs = 0.5f * fmaxf(At[cur][pl][ql] + BtT[cur][pl][ql], 0.0f);
      for (int bk = 0; bk < 6; ++bk)
        prod[bk] *= 1.0f - cs * wch[cur][bk*64 + pl] * sreg[bk];
    }
    __syncthreads();
  }

  for (int bk = 0; bk < 6; ++bk) red[bk][psub][ql] = prod[bk];
  __syncthreads();
  if (t < 64) {
    for (int bk = 0; bk < 6; ++bk) {
      const float P = red[bk][0][ql] * red[bk][1][ql] * red[bk][2][ql] * red[bk][3][ql];
      part[(size_t)pseg*BKN + bk*NN + q0 + ql] = P;
    }
  }
}

// ---------------------------------------------------------------------------
// Phase 3: reduce p-segment partials, substitute p2, newState, newI
// ---------------------------------------------------------------------------
__global__ void finalize_kernel(const float* __restrict__ state,
                                const float* __restrict__ psts,
                                const float* __restrict__ part,  // [pseg][g]
                                float* __restrict__ out) {
  const int g = blockIdx.x * blockDim.x + threadIdx.x;
  if (g >= BKN) return;
  const float* sp = state + (size_t)g * TSZ;
  const float st0 = sp[0], st1 = sp[1], st2 = sp[2], st3 = sp[3];
  float P[16];
  const float* pp = psts + (size_t)g * 16;
  for (int r = 0; r < 16; ++r) P[r] = pp[r];
  float prodP = 1.0f;
  for (int s = 0; s < PSPLIT; ++s) prodP *= part[(size_t)s*BKN + g];
  const float p2 = (1.0f - prodP) / (st0 + 1e-8f);
  P[4] = p2;           // (c=1,a=0)
  P[0] = 1.0f - p2;    // (c=0,a=0)
  const float newI = P[9] * st1;   // (c=2,a=1) * e
  float* ns = out + (size_t)g * TSZ;
  for (int c = 0; c < 4; ++c)
    ns[c] = P[c*4+0]*st0 + P[c*4+1]*st1 + P[c*4+2]*st2 + P[c*4+3]*st3;
  out[(size_t)BKN*TSZ + g] = newI;
}

// ---------------------------------------------------------------------------
extern "C" void kernel_launch(void* const* d_in, const int* in_sizes, int n_in,
                              void* d_out, int out_size, void* d_ws, size_t ws_size,
                              hipStream_t stream) {
  const float* state   = (const float*)d_in[0];
  const float* tt      = (const float*)d_in[1];
  const float* contact = (const float*)d_in[2];
  const float* ps      = (const float*)d_in[3];
  const float* cim     = (const float*)d_in[4];
  const float* Ts      = (const float*)d_in[5];
  const float* trust   = (const float*)d_in[6];
  const int*   cat     = (const int*)d_in[7];
  float* ws  = (float*)d_ws;
  float* out = (float*)d_out;

  prep_kernel<<<1, 32, 0, stream>>>(ps, cim, Ts, trust, tt, cat, ws + WS_LINC);
  psts_kernel<<<BB*(NN/16), 32, 0, stream>>>(state, ws + WS_LINC,
                                             ws + WS_PSTS, ws + WS_W);
  prod_kernel<<<64*PSPLIT, 256, 0, stream>>>(contact, state, ws + WS_W,
                                             ws + WS_PART);
  finalize_kernel<<<(BKN + 255)/256, 256, 0, stream>>>(state, ws + WS_PSTS,
                                                       ws + WS_PART, out);
}